// EulerLoss_80479097192949
// MI455X (gfx1250) — compile-verified
//
#include <hip/hip_runtime.h>
#include <hip/hip_bf16.h>
#include <math.h>

#define BLOCK 256
#define NBLK  1024

typedef __attribute__((ext_vector_type(2))) float v2f;
typedef __attribute__((ext_vector_type(8))) float v8f;

// ---------------------------------------------------------------------------
// Wave-wide sum via V_WMMA_F32_16X16X4_F32.
// Each lane places its value into a unique (M,K) slot of the 16x4 A matrix
// (A VGPR0 = value, A VGPR1 = 0). With B = all-ones, D[m][n] = sum_k A[m][k],
// so the 16 row sums partition the 32 lane values. Lane L holds the 8 row
// sums for M in {0..7} (L<16) or {8..15} (L>=16) in its 8 D VGPRs; summing
// them and adding the xor-16 partner gives the full wave sum in every lane,
// in full f32 precision. Requires EXEC == all ones (callers guarantee it).
// ---------------------------------------------------------------------------
__device__ __forceinline__ float wave_sum_wmma(float v) {
    v2f a;    a[0] = v;    a[1] = 0.0f;
    v2f ones; ones[0] = 1.0f; ones[1] = 1.0f;
    v8f c = {};
    v8f d = __builtin_amdgcn_wmma_f32_16x16x4_f32(
        /*neg_a=*/false, a, /*neg_b=*/false, ones,
        /*c_mod=*/(short)0, c, /*reuse_a=*/false, /*reuse_b=*/false);
    float s = ((d[0] + d[1]) + (d[2] + d[3])) + ((d[4] + d[5]) + (d[6] + d[7]));
    s += __shfl_xor(s, 16, 32);   // combine M 0..7 half with M 8..15 half
    return s;
}

__device__ __forceinline__ float sl1(float x) {
    float d = fabsf(x);
    // beta = 0.01 ; 0.5/beta = 50
    return d < 0.01f ? 50.0f * d * d : d - 0.005f;
}

__device__ __forceinline__ void quat2mat(float r, float i, float j, float k,
                                         float* __restrict__ m) {
    float s = 2.0f / (r * r + i * i + j * j + k * k);
    m[0] = 1.0f - s * (j * j + k * k);
    m[1] = s * (i * j - k * r);
    m[2] = s * (i * k + j * r);
    m[3] = s * (i * j + k * r);
    m[4] = 1.0f - s * (i * i + k * k);
    m[5] = s * (j * k - i * r);
    m[6] = s * (i * k - j * r);
    m[7] = s * (j * k + i * r);
    m[8] = 1.0f - s * (i * i + j * j);
}

// Faithful 3x3 inverse (adjugate / det), matching jnp.linalg.inv semantics.
__device__ __forceinline__ void inv3(const float* __restrict__ m,
                                     float* __restrict__ o) {
    float c00 = m[4] * m[8] - m[5] * m[7];
    float c01 = m[5] * m[6] - m[3] * m[8];
    float c02 = m[3] * m[7] - m[4] * m[6];
    float det = m[0] * c00 + m[1] * c01 + m[2] * c02;
    float id  = 1.0f / det;
    o[0] = c00 * id;
    o[1] = (m[2] * m[7] - m[1] * m[8]) * id;
    o[2] = (m[1] * m[5] - m[2] * m[4]) * id;
    o[3] = c01 * id;
    o[4] = (m[0] * m[8] - m[2] * m[6]) * id;
    o[5] = (m[2] * m[3] - m[0] * m[5]) * id;
    o[6] = c02 * id;
    o[7] = (m[1] * m[6] - m[0] * m[7]) * id;
    o[8] = (m[0] * m[4] - m[1] * m[3]) * id;
}

// ---------------------------------------------------------------------------
// Pass 1: grid-stride over samples, per-thread f32 partials, WMMA wave sum,
// LDS across the 8 waves, one deterministic partial pair per block.
// ---------------------------------------------------------------------------
__global__ __launch_bounds__(BLOCK) void el_partial(
    const float* __restrict__ tt,   // target_transl (b,3)
    const float* __restrict__ tr,   // target_rot    (b,4)
    const float* __restrict__ te,   // transl_err    (b,3)
    const float* __restrict__ re,   // rot_err       (b,4)
    float* __restrict__ part, int b) {
    const float4* __restrict__ tr4 = (const float4*)tr;
    const float4* __restrict__ re4 = (const float4*)re;

    float lr = 0.0f, lt = 0.0f;
    for (int i = blockIdx.x * BLOCK + threadIdx.x; i < b;
         i += gridDim.x * BLOCK) {
        float4 q = tr4[i];                         // global_load_b128
        float4 p = re4[i];                         // global_load_b128
        float t0 = tt[3 * i], t1 = tt[3 * i + 1], t2 = tt[3 * i + 2];
        float e0 = te[3 * i], e1 = te[3 * i + 1], e2 = te[3 * i + 2];

        float R[9], Ri[9], P[9];
        quat2mat(q.x, q.y, q.z, q.w, R);
        inv3(R, Ri);
        quat2mat(p.x, p.y, p.z, p.w, P);

        // T_inv = -R_inv @ target_transl
        float Ti0 = -(Ri[0] * t0 + Ri[1] * t1 + Ri[2] * t2);
        float Ti1 = -(Ri[3] * t0 + Ri[4] * t1 + Ri[5] * t2);
        float Ti2 = -(Ri[6] * t0 + Ri[7] * t1 + Ri[8] * t2);

        // rotation loss: smooth_l1(R_inv @ P - I)
#pragma unroll
        for (int r = 0; r < 3; ++r) {
#pragma unroll
            for (int cc = 0; cc < 3; ++cc) {
                float mrc = Ri[3 * r + 0] * P[cc] +
                            Ri[3 * r + 1] * P[3 + cc] +
                            Ri[3 * r + 2] * P[6 + cc];
                lr += sl1(mrc - (r == cc ? 1.0f : 0.0f));
            }
        }

        // translation loss: smooth_l1(P @ T_inv + transl_err)
        float m0 = P[0] * Ti0 + P[1] * Ti1 + P[2] * Ti2 + e0;
        float m1 = P[3] * Ti0 + P[4] * Ti1 + P[5] * Ti2 + e1;
        float m2 = P[6] * Ti0 + P[7] * Ti1 + P[8] * Ti2 + e2;
        lt += sl1(m0) + sl1(m1) + sl1(m2);
    }

    // wave reduction via WMMA (all 256 threads active -> EXEC all ones)
    float wlr = wave_sum_wmma(lr);
    float wlt = wave_sum_wmma(lt);

    __shared__ float sr[BLOCK / 32], st[BLOCK / 32];
    int wv = threadIdx.x >> 5;
    if ((threadIdx.x & 31) == 0) { sr[wv] = wlr; st[wv] = wlt; }
    __syncthreads();
    if (threadIdx.x == 0) {
        float r = 0.0f, t = 0.0f;
#pragma unroll
        for (int w = 0; w < BLOCK / 32; ++w) { r += sr[w]; t += st[w]; }
        part[2 * blockIdx.x]     = r;
        part[2 * blockIdx.x + 1] = t;
    }
}

// ---------------------------------------------------------------------------
// Pass 2: single 1024-thread block (32 waves, EXEC all ones) folds the NBLK
// per-block partials deterministically and emits [total, loss_t, loss_r].
// ---------------------------------------------------------------------------
__global__ __launch_bounds__(NBLK) void el_final(
    const float* __restrict__ part, float* __restrict__ out, float invb) {
    float lr = part[2 * threadIdx.x];
    float lt = part[2 * threadIdx.x + 1];

    float wlr = wave_sum_wmma(lr);
    float wlt = wave_sum_wmma(lt);

    __shared__ float sr[NBLK / 32], st[NBLK / 32];
    int wv = threadIdx.x >> 5;
    if ((threadIdx.x & 31) == 0) { sr[wv] = wlr; st[wv] = wlt; }
    __syncthreads();
    if (threadIdx.x == 0) {
        float r = 0.0f, t = 0.0f;
#pragma unroll
        for (int w = 0; w < NBLK / 32; ++w) { r += sr[w]; t += st[w]; }
        r *= invb;
        t *= invb;
        out[0] = r + t;   // total_loss
        out[1] = t;       // loss_t
        out[2] = r;       // loss_r
    }
}

extern "C" void kernel_launch(void* const* d_in, const int* in_sizes, int n_in,
                              void* d_out, int out_size, void* d_ws,
                              size_t ws_size, hipStream_t stream) {
    // d_in[0] = point_clouds (unused by the reference computation)
    const float* tt = (const float*)d_in[1];
    const float* tr = (const float*)d_in[2];
    const float* te = (const float*)d_in[3];
    const float* re = (const float*)d_in[4];
    int b = in_sizes[1] / 3;

    float* part = (float*)d_ws;   // NBLK * 2 floats = 8 KB scratch

    el_partial<<<NBLK, BLOCK, 0, stream>>>(tt, tr, te, re, part, b);
    el_final<<<1, NBLK, 0, stream>>>(part, (float*)d_out, 1.0f / (float)b);
}